// HMM_59691455480213
// MI455X (gfx1250) — compile-verified
//
#include <hip/hip_runtime.h>
#include <math.h>

// ---------------------------------------------------------------------------
// HMM forward (log space) on gfx1250.
//   a_t = logsumexp_j(a_{t-1}[j] + logA[:,j]) + e_t
// rewritten as scaled forward:
//   M = max(a_prev); p = exp(a_prev - M); s = A @ p; a_t = log(s) + M + e_t
// The GEMV s = A @ p uses V_WMMA_F32_16X16X4_F32: 16x4 A tile is real data,
// 4x16 B tile is the p-chunk broadcast to all 16 columns, so column 0 of the
// 16x16 accumulator is the GEMV partial sum.
// Persistent kernel: 64 WGs x 256 threads, one 16-row strip of A per WG kept
// normalized in LDS; grid barrier per time step via monotonic atomic counter.
// ---------------------------------------------------------------------------

#define NZ 1024
#define NX 4096
#define NWG 64
#define ROWS_PER_WG 16   // NZ / NWG
#define TPB 256          // 8 wave32s
#define NWAVE 8
#define ASTRIDE 1028     // padded LDS row stride (floats): 4*m bank stride -> conflict-free

typedef float v2f __attribute__((ext_vector_type(2)));
typedef float v8f __attribute__((ext_vector_type(8)));

__device__ __forceinline__ float wave_reduce_sum(float v) {
    #pragma unroll
    for (int off = 16; off > 0; off >>= 1) v += __shfl_down(v, off, 32);
    return __shfl(v, 0, 32);
}

__device__ __forceinline__ float wave_reduce_max(float v) {
    #pragma unroll
    for (int off = 16; off > 0; off >>= 1) v = fmaxf(v, __shfl_down(v, off, 32));
    return v;  // valid in lane 0
}

// ---- pre-pass 1: logSumB[z] = log(sum_x emission[z,x]) ---------------------
__global__ __launch_bounds__(TPB) void row_logsum_kernel(const float* __restrict__ emission,
                                                         float* __restrict__ logSumB) {
    __shared__ float wsum[NWAVE];
    const int z = blockIdx.x;
    const float* row = emission + (size_t)z * NX;
    float s = 0.f;
    for (int x = threadIdx.x; x < NX; x += TPB) s += row[x];
    s = wave_reduce_sum(s);
    const int w = threadIdx.x >> 5, lane = threadIdx.x & 31;
    if (lane == 0) wsum[w] = s;
    __syncthreads();
    if (threadIdx.x == 0) {
        float t = 0.f;
        #pragma unroll
        for (int i = 0; i < NWAVE; ++i) t += wsum[i];
        logSumB[z] = logf(t);
    }
}

// ---- pre-pass 2: a0[z] = log(pi[z]) + log B[z, obs0] -----------------------
__global__ __launch_bounds__(TPB) void init_kernel(const float* __restrict__ start_prob,
                                                   const float* __restrict__ emission,
                                                   const int* __restrict__ obs,
                                                   const float* __restrict__ logSumB,
                                                   float* __restrict__ alpha,
                                                   float* __restrict__ abuf0) {
    __shared__ float red[TPB];
    float x[4];
    float s = 0.f;
    #pragma unroll
    for (int i = 0; i < 4; ++i) { x[i] = start_prob[threadIdx.x + TPB * i]; s += x[i]; }
    red[threadIdx.x] = s;
    __syncthreads();
    for (int o = TPB / 2; o > 0; o >>= 1) {
        if (threadIdx.x < o) red[threadIdx.x] += red[threadIdx.x + o];
        __syncthreads();
    }
    const float logS = logf(red[0]);
    const int o0 = obs[0];
    #pragma unroll
    for (int i = 0; i < 4; ++i) {
        const int z = threadIdx.x + TPB * i;
        const float a = logf(x[i]) - logS + logf(emission[(size_t)z * NX + o0]) - logSumB[z];
        abuf0[z] = a;
        alpha[(size_t)z * NX + 0] = a;
    }
}

// ---- main persistent scan kernel -------------------------------------------
__global__ __launch_bounds__(TPB) void hmm_forward_kernel(const float* __restrict__ transition,
                                                          const float* __restrict__ emission,
                                                          const int* __restrict__ obs,
                                                          const float* __restrict__ logSumB,
                                                          float* __restrict__ alpha,
                                                          float* __restrict__ abuf0,
                                                          float* __restrict__ abuf1,
                                                          unsigned* __restrict__ cnt) {
    extern __shared__ float smem[];
    float* As   = smem;                          // ROWS_PER_WG * ASTRIDE floats (~64KB)
    float* p    = As + ROWS_PER_WG * ASTRIDE;    // NZ floats
    float* red8 = p + NZ;                        // NWAVE floats (wave maxes)
    float* part = red8 + NWAVE;                  // NWAVE * 16 floats

    const int tid  = threadIdx.x;
    const int lane = tid & 31;
    const int w    = tid >> 5;
    const int r0   = blockIdx.x * ROWS_PER_WG;

    // Stage this WG's 16 rows of A into LDS, row-normalized (one time).
    for (int rr = 0; rr < 2; ++rr) {
        const int r = w * 2 + rr;
        const float* trow = transition + (size_t)(r0 + r) * NZ;
        float s = 0.f;
        for (int j = lane; j < NZ; j += 32) s += trow[j];
        s = wave_reduce_sum(s);
        const float inv = 1.0f / s;
        for (int j = lane; j < NZ; j += 32) As[r * ASTRIDE + j] = trow[j] * inv;
    }
    __syncthreads();

    // WMMA f32 16x16x4 operand addressing (A-matrix VGPR layout):
    //   lanes 0-15 : M=lane,    VGPR0=K0, VGPR1=K1
    //   lanes 16-31: M=lane-16, VGPR0=K2, VGPR1=K3
    const int m  = lane & 15;
    const int ko = (lane < 16) ? 0 : 2;
    const float* Arow = As + m * ASTRIDE;

    // Loop-invariant per-output-row state (tid < ROWS_PER_WG only).
    const bool  isOut = (tid < ROWS_PER_WG);
    const int   zOut  = r0 + tid;
    const float lsb   = isOut ? logSumB[zOut] : 0.f;
    const float* erow = emission + (size_t)zOut * NX;
    float* arow = alpha + (size_t)zOut * NX;

    for (int t = 1; t < NX; ++t) {
        // ---- grid barrier: all WGs have published a_{t-1} ----
        __syncthreads();
        if (tid == 0) {
            __threadfence();
            __hip_atomic_fetch_add(cnt, 1u, __ATOMIC_ACQ_REL, __HIP_MEMORY_SCOPE_AGENT);
            const unsigned target = (unsigned)t * NWG;   // monotonic counter, no reset race
            while (__hip_atomic_load(cnt, __ATOMIC_ACQUIRE, __HIP_MEMORY_SCOPE_AGENT) < target)
                __builtin_amdgcn_s_sleep(1);
            __threadfence();
        }
        __syncthreads();

        // ---- prefetch e_t inputs early: overlaps L2 latency with GEMV ----
        float ev = 0.f;
        if (isOut) ev = erow[obs[t]];

        const float* aprev = (t & 1) ? abuf0 : abuf1;
        float*       anext = (t & 1) ? abuf1 : abuf0;

        // ---- M = max(a_prev): wave shuffle reduce + one 8-entry LDS stage ----
        const float x0 = aprev[tid], x1 = aprev[tid + 256], x2 = aprev[tid + 512], x3 = aprev[tid + 768];
        const float wm = wave_reduce_max(fmaxf(fmaxf(x0, x1), fmaxf(x2, x3)));
        if (lane == 0) red8[w] = wm;
        __syncthreads();
        float M = red8[0];
        #pragma unroll
        for (int i = 1; i < NWAVE; ++i) M = fmaxf(M, red8[i]);

        // ---- p = exp(a_prev - M) into LDS ----
        p[tid]       = expf(x0 - M);
        p[tid + 256] = expf(x1 - M);
        p[tid + 512] = expf(x2 - M);
        p[tid + 768] = expf(x3 - M);
        __syncthreads();

        // ---- s = A_strip @ p via WMMA; each wave covers a K-block of 128 ----
        v8f c = {0.f, 0.f, 0.f, 0.f, 0.f, 0.f, 0.f, 0.f};
        int kb = w * 128 + ko;
        #pragma unroll 4
        for (int kk = 0; kk < 32; ++kk, kb += 4) {
            v2f a = *(const v2f*)(Arow + kb);   // 16x4 A tile (ds_load_2addr_b64, conflict-free)
            v2f b = *(const v2f*)(p + kb);      // 4x16 B tile: p chunk broadcast across columns
            c = __builtin_amdgcn_wmma_f32_16x16x4_f32(false, a, false, b, (short)0, c,
                                                      false, false);
        }
        // Column 0 of the 16x16 accumulator: lane 0 holds rows 0-7, lane 16 rows 8-15.
        if ((lane & 15) == 0) {
            float* dst = part + w * 16 + ((lane == 0) ? 0 : 8);
            #pragma unroll
            for (int v = 0; v < 8; ++v) dst[v] = c[v];
        }
        __syncthreads();

        // ---- a_t = log(s) + M + e_t ; publish ----
        if (isOut) {
            float s = 0.f;
            #pragma unroll
            for (int ww = 0; ww < NWAVE; ++ww) s += part[ww * 16 + tid];
            const float a = logf(s) + M + (logf(ev) - lsb);
            anext[zOut] = a;
            arow[t] = a;
            __threadfence();
        }
    }
}

// ---------------------------------------------------------------------------
extern "C" void kernel_launch(void* const* d_in, const int* in_sizes, int n_in,
                              void* d_out, int out_size, void* d_ws, size_t ws_size,
                              hipStream_t stream) {
    const int*   obs        = (const int*)  d_in[0];
    const float* start_prob = (const float*)d_in[1];
    const float* transition = (const float*)d_in[2];
    const float* emission   = (const float*)d_in[3];
    float* alpha = (float*)d_out;

    char* ws = (char*)d_ws;
    unsigned* cnt    = (unsigned*)ws;                 // barrier counter (zeroed each launch)
    float*    abuf0  = (float*)(ws + 256);            // a_t double buffer 0
    float*    abuf1  = (float*)(ws + 256 + NZ * 4);   // a_t double buffer 1
    float*    logSumB= (float*)(ws + 256 + 2 * NZ * 4);

    hipMemsetAsync(ws, 0, 256, stream);  // reset barrier counter (graph-capture safe)

    row_logsum_kernel<<<NZ, TPB, 0, stream>>>(emission, logSumB);
    init_kernel<<<1, TPB, 0, stream>>>(start_prob, emission, obs, logSumB, alpha, abuf0);

    const size_t shmem = (size_t)(ROWS_PER_WG * ASTRIDE + NZ + NWAVE + NWAVE * 16) * sizeof(float);
    hmm_forward_kernel<<<NWG, TPB, shmem, stream>>>(transition, emission, obs, logSumB,
                                                    alpha, abuf0, abuf1, cnt);
}